// HeteroGNN_57449482551719
// MI455X (gfx1250) — compile-verified
//
#include <hip/hip_runtime.h>
#include <hip/hip_bf16.h>
#include <math.h>

// ---------------------------------------------------------------------------
// HeteroGNN (2x GCN layers, 4 relations w/ shared weights, graph-LayerNorm)
// for MI455X (gfx1250, wave32).
//
//   layer: h = x @ W (WMMA f32 16x16x4)          -> hbuf
//          agg[dst] += ew * h[src]  (4 relations) -> d_out (L2-resident atomics)
//          stats = sum/sumsq(agg + b)             -> f64 reduction
//          out = (agg + b - m)/(std+eps)*lw + lb  -> d_out in place
// ---------------------------------------------------------------------------

typedef __attribute__((ext_vector_type(2))) float v2f;
typedef __attribute__((ext_vector_type(8))) float v8f;

#define HDIM 256
#define EPSV 1e-5f

// ---------------------------------------------------------------------------
// GEMM: C[M x 256] = A[M x K] * B[K x 256], f32, via V_WMMA_F32_16X16X4_F32.
// One wave computes a 16x64 output strip (4 accumulators, A fragment reused).
// M must be a multiple of 16 (N = 100000 = 6250*16). K multiple of 4.
//
// A frag (16x4 f32, ISA 7.12.2): lanes 0-15 row M=lane hold {K=k0, k0+1},
//                                lanes 16-31 row M=lane-16 hold {K=k0+2, k0+3}.
// B frag (4x16): VGPR0 = rows k0 (lanes 0-15) / k0+2 (lanes 16-31),
//                VGPR1 = rows k0+1 / k0+3; column = lane & 15.
// C/D (16x16 f32, 8 VGPRs): VGPR i -> row m0+i (lanes 0-15), m0+i+8 (16-31).
// ---------------------------------------------------------------------------
__global__ __launch_bounds__(256) void gemm_wmma_f32(
    const float* __restrict__ A, const float* __restrict__ B,
    float* __restrict__ C, int M, int K)
{
    const int wave  = (int)((blockIdx.x * blockDim.x + threadIdx.x) >> 5);
    const int lane  = (int)(threadIdx.x & 31);
    const int strip = wave & 3;            // 4 strips of 64 columns
    const int m0    = (wave >> 2) << 4;    // 16 rows per wave
    if (m0 >= M) return;                   // uniform across the wave
    const int n0    = strip << 6;
    const int half  = lane >> 4;           // 0: lanes 0-15, 1: lanes 16-31
    const int r     = lane & 15;

    v8f c0 = {}, c1 = {}, c2 = {}, c3 = {};

    const float* __restrict__ arow = A + (size_t)(m0 + r) * (size_t)K;

    for (int k0 = 0; k0 < K; k0 += 4) {
        v2f a;
        a.x = arow[k0 + 2 * half];
        a.y = arow[k0 + 2 * half + 1];

        const float* __restrict__ brow0 = B + (size_t)(k0 + 2 * half) * HDIM;
        const float* __restrict__ brow1 = brow0 + HDIM;
        const int cb = n0 + r;

        v2f b;
        b.x = brow0[cb];      b.y = brow1[cb];
        c0 = __builtin_amdgcn_wmma_f32_16x16x4_f32(false, a, false, b, (short)0, c0, false, false);
        b.x = brow0[cb + 16]; b.y = brow1[cb + 16];
        c1 = __builtin_amdgcn_wmma_f32_16x16x4_f32(false, a, false, b, (short)0, c1, false, false);
        b.x = brow0[cb + 32]; b.y = brow1[cb + 32];
        c2 = __builtin_amdgcn_wmma_f32_16x16x4_f32(false, a, false, b, (short)0, c2, false, false);
        b.x = brow0[cb + 48]; b.y = brow1[cb + 48];
        c3 = __builtin_amdgcn_wmma_f32_16x16x4_f32(false, a, false, b, (short)0, c3, false, false);
    }

    float* __restrict__ cbase = C + (size_t)(m0 + 8 * half) * HDIM + n0 + r;
#pragma unroll
    for (int i = 0; i < 8; ++i) {
        cbase[(size_t)i * HDIM +  0] = c0[i];
        cbase[(size_t)i * HDIM + 16] = c1[i];
        cbase[(size_t)i * HDIM + 32] = c2[i];
        cbase[(size_t)i * HDIM + 48] = c3[i];
    }
}

// ---------------------------------------------------------------------------
// Zero the aggregation buffer + the f64 reduction cells.
// ---------------------------------------------------------------------------
__global__ __launch_bounds__(256) void zero_kernel(float* __restrict__ p, int n,
                                                   double* __restrict__ red)
{
    int i      = (int)(blockIdx.x * blockDim.x + threadIdx.x);
    int stride = (int)(gridDim.x * blockDim.x);
    if (i < 4) red[i] = 0.0;
    for (; i < n; i += stride) p[i] = 0.0f;
}

// ---------------------------------------------------------------------------
// Edge scatter: one wave32 per edge. Lane l handles 8 contiguous floats
// (two coalesced float4 gathers from h[src], 8 hw fp32 atomics into out[dst]).
// h (102 MB) is L2-resident on MI455X (192 MB L2), so gathers hit L2.
// ---------------------------------------------------------------------------
__global__ __launch_bounds__(256) void scatter_kernel(
    const float* __restrict__ h, const int* __restrict__ ei,
    const float* __restrict__ ew, float* __restrict__ out, int E)
{
    const int lane   = (int)(threadIdx.x & 31);
    int       e      = (int)((blockIdx.x * blockDim.x + threadIdx.x) >> 5);
    const int nwaves = (int)((gridDim.x * blockDim.x) >> 5);

    for (; e < E; e += nwaves) {
        const int   src = ei[e];        // row 0 of (2,E)
        const int   dst = ei[E + e];    // row 1 of (2,E)
        const float w   = ew[e];

        const float4* __restrict__ hp =
            (const float4*)(h + (size_t)src * HDIM) + (lane << 1);
        const float4 v0 = hp[0];
        const float4 v1 = hp[1];

        float* __restrict__ op = out + (size_t)dst * HDIM + (lane << 3);
        unsafeAtomicAdd(op + 0, w * v0.x);
        unsafeAtomicAdd(op + 1, w * v0.y);
        unsafeAtomicAdd(op + 2, w * v0.z);
        unsafeAtomicAdd(op + 3, w * v0.w);
        unsafeAtomicAdd(op + 4, w * v1.x);
        unsafeAtomicAdd(op + 5, w * v1.y);
        unsafeAtomicAdd(op + 6, w * v1.z);
        unsafeAtomicAdd(op + 7, w * v1.w);
    }
}

// ---------------------------------------------------------------------------
// Global sum / sum-of-squares of (agg + bias[col]) in f64.
// ---------------------------------------------------------------------------
__global__ __launch_bounds__(256) void reduce_kernel(
    const float* __restrict__ agg, const float* __restrict__ bias,
    double* __restrict__ red, int n)
{
    __shared__ double sdata[2][8];
    double s = 0.0, s2 = 0.0;

    int i      = (int)(blockIdx.x * blockDim.x + threadIdx.x);
    int stride = (int)(gridDim.x * blockDim.x);
    for (; i < n; i += stride) {
        double v = (double)agg[i] + (double)bias[i & (HDIM - 1)];
        s += v;
        s2 += v * v;
    }
#pragma unroll
    for (int off = 16; off > 0; off >>= 1) {
        s  += __shfl_down(s,  off, 32);
        s2 += __shfl_down(s2, off, 32);
    }
    const int lane = (int)(threadIdx.x & 31);
    const int w    = (int)(threadIdx.x >> 5);
    if (lane == 0) { sdata[0][w] = s; sdata[1][w] = s2; }
    __syncthreads();
    if (threadIdx.x == 0) {
        double ts = 0.0, ts2 = 0.0;
#pragma unroll
        for (int k = 0; k < 8; ++k) { ts += sdata[0][k]; ts2 += sdata[1][k]; }
        unsafeAtomicAdd(&red[0], ts);
        unsafeAtomicAdd(&red[1], ts2);
    }
}

// ---------------------------------------------------------------------------
// Fused bias + graph-LayerNorm affine (in place allowed: elementwise).
// ---------------------------------------------------------------------------
__global__ __launch_bounds__(256) void norm_kernel(
    const float* __restrict__ agg, const float* __restrict__ bias,
    const float* __restrict__ lw, const float* __restrict__ lb,
    const double* __restrict__ red, float* __restrict__ out, int n)
{
    const double cnt = (double)n;
    const double m   = red[0] / cnt;
    double       var = red[1] / cnt - m * m;
    if (var < 0.0) var = 0.0;
    const float stdv = sqrtf((float)var);
    const float inv  = 1.0f / (stdv + EPSV);
    const float mf   = (float)m;

    int i      = (int)(blockIdx.x * blockDim.x + threadIdx.x);
    int stride = (int)(gridDim.x * blockDim.x);
    for (; i < n; i += stride) {
        const int col = i & (HDIM - 1);
        out[i] = (agg[i] + bias[col] - mf) * inv * lw[col] + lb[col];
    }
}

// ---------------------------------------------------------------------------
extern "C" void kernel_launch(void* const* d_in, const int* in_sizes, int n_in,
                              void* d_out, int out_size, void* d_ws, size_t ws_size,
                              hipStream_t stream)
{
    const float* x  = (const float*)d_in[0];
    const int*   ei[4] = { (const int*)d_in[1], (const int*)d_in[3],
                           (const int*)d_in[5], (const int*)d_in[7] };
    const float* ew[4] = { (const float*)d_in[2], (const float*)d_in[4],
                           (const float*)d_in[6], (const float*)d_in[8] };
    const int    Ecnt[4] = { in_sizes[2], in_sizes[4], in_sizes[6], in_sizes[8] };
    const float* W1   = (const float*)d_in[9];
    const float* b1   = (const float*)d_in[10];
    const float* W2   = (const float*)d_in[11];
    const float* b2   = (const float*)d_in[12];
    const float* ln1w = (const float*)d_in[13];
    const float* ln1b = (const float*)d_in[14];
    const float* ln2w = (const float*)d_in[15];
    const float* ln2b = (const float*)d_in[16];

    float* out = (float*)d_out;

    const int NH = out_size;          // N * 256
    const int Nn = NH / HDIM;         // 100000
    const int K1 = in_sizes[0] / Nn;  // 128

    // Workspace: [0,256) f64 reduction cells; then N*H floats of GEMM output.
    double* red  = (double*)d_ws;
    float*  hbuf = (float*)((char*)d_ws + 256);

    const int gemmWaves  = (Nn / 16) * 4;                 // 25000
    const int gemmBlocks = (gemmWaves + 7) / 8;           // 8 waves / block
    const int elemBlocks = 4096;                          // grid-stride passes

    // ---------------- layer 1 ----------------
    gemm_wmma_f32<<<gemmBlocks, 256, 0, stream>>>(x, W1, hbuf, Nn, K1);
    zero_kernel<<<elemBlocks, 256, 0, stream>>>(out, NH, red);
    for (int r = 0; r < 4; ++r) {
        int blocks = (Ecnt[r] + 7) / 8;                   // 8 edges (waves)/block
        scatter_kernel<<<blocks, 256, 0, stream>>>(hbuf, ei[r], ew[r], out, Ecnt[r]);
    }
    reduce_kernel<<<elemBlocks, 256, 0, stream>>>(out, b1, red, NH);
    norm_kernel<<<elemBlocks, 256, 0, stream>>>(out, b1, ln1w, ln1b, red, out, NH);

    // ---------------- layer 2 ----------------
    gemm_wmma_f32<<<gemmBlocks, 256, 0, stream>>>(out, W2, hbuf, Nn, HDIM);
    zero_kernel<<<elemBlocks, 256, 0, stream>>>(out, NH, red);
    for (int r = 0; r < 4; ++r) {
        int blocks = (Ecnt[r] + 7) / 8;
        scatter_kernel<<<blocks, 256, 0, stream>>>(hbuf, ei[r], ew[r], out, Ecnt[r]);
    }
    reduce_kernel<<<elemBlocks, 256, 0, stream>>>(out, b2, red, NH);
    norm_kernel<<<elemBlocks, 256, 0, stream>>>(out, b2, ln2w, ln2b, red, out, NH);
}